// GraphEncoder_49761491091473
// MI455X (gfx1250) — compile-verified
//
#include <hip/hip_runtime.h>
#include <math.h>

#define DOUT 64
#define NEG_SLOPE 0.2f
#define SM_EPS 1e-16f
#define BN_EPS 1e-5f

typedef __attribute__((ext_vector_type(2))) float v2f;
typedef __attribute__((ext_vector_type(8))) float v8f;

// ---------------------------------------------------------------------------
// GEMM H[N,64] = X[N,K] @ W[K,64] via V_WMMA_F32_16X16X4_F32 (fp32 matrix op).
// K compile-time (10 or 64). Full chunks: unguarded clause'd loads. Tail
// (K=10): unconditional clamped loads + multiply-mask zeroing (no exec math).
// One block = 16 output rows, 4 waves; wave w computes col tile n0=16w.
// A frag (16x4 f32, 2 VGPR/lane): lanes 0-15 hold K=kc+0/kc+1 for row M=lane,
// lanes 16-31 hold K=kc+2/kc+3 (one 8B load: K even & ka even => 8B aligned).
// C/D (16x16 f32, 8 VGPR): vgpr r -> row r (lanes 0-15) / row r+8 (lanes 16-31).
// ---------------------------------------------------------------------------
template <int K>
__global__ __launch_bounds__(128) void gemm_wmma_f32(
    const float* __restrict__ X, const float* __restrict__ W,
    float* __restrict__ H, int Nn)
{
  static_assert((K & 1) == 0, "K must be even for aligned v2f A-loads");
  const int lane   = threadIdx.x & 31;
  const int wave   = threadIdx.x >> 5;
  const int n0     = wave * 16;
  const int laneLo = lane & 15;
  const bool hi    = lane >= 16;
  const int mBase  = blockIdx.x * 16;

  int row = mBase + laneLo;
  if (row >= Nn) row = Nn - 1;  // clamp: OOB rows compute garbage, never stored
  const float* Xr = X + (size_t)row * K;
  const float* Wc = W + n0 + laneLo;

  v8f c = {0.f, 0.f, 0.f, 0.f, 0.f, 0.f, 0.f, 0.f};

  constexpr int KF = K & ~3;  // full 4-wide chunks
#pragma unroll
  for (int kc = 0; kc < KF; kc += 4) {
    const int ka = kc + (hi ? 2 : 0);
    const v2f a = *(const v2f*)(Xr + ka);  // 8B aligned (K even, ka even)
    v2f b;
    b.x = Wc[(size_t)ka * DOUT];
    b.y = Wc[(size_t)(ka + 1) * DOUT];
    c = __builtin_amdgcn_wmma_f32_16x16x4_f32(
        false, a, false, b, (short)0, c, false, false);
  }

  if constexpr (KF < K) {
    // Tail chunk (K=10 -> ks {8,9} valid, {10,11} zero). Loads are
    // unconditional at clamped in-range addresses; zeroing is done with a
    // multiply mask so the compiler emits load + v_cndmask, not exec branches.
    const int ka   = KF + (hi ? 2 : 0);
    const int ia0  = (ka     < K) ? ka     : (K - 1);
    const int ia1  = (ka + 1 < K) ? ka + 1 : (K - 1);
    const float m0 = (ka     < K) ? 1.f : 0.f;
    const float m1 = (ka + 1 < K) ? 1.f : 0.f;
    v2f a, b;
    a.x = Xr[ia0] * m0;
    a.y = Xr[ia1] * m1;
    b.x = Wc[(size_t)ia0 * DOUT] * m0;
    b.y = Wc[(size_t)ia1 * DOUT] * m1;
    c = __builtin_amdgcn_wmma_f32_16x16x4_f32(
        false, a, false, b, (short)0, c, false, false);
  }

  // Store: block-level range test; hot path is 8 unguarded coalesced stores.
  const int rowBase = mBase + (hi ? 8 : 0);
  float* Hp = H + (size_t)rowBase * DOUT + n0 + laneLo;
  if (mBase + 16 <= Nn) {
#pragma unroll
    for (int r = 0; r < 8; ++r) Hp[(size_t)r * DOUT] = c[r];
  } else {
#pragma unroll
    for (int r = 0; r < 8; ++r) {
      if (rowBase + r < Nn) Hp[(size_t)r * DOUT] = c[r];
    }
  }
}

// ---- order-preserving float<->uint encoding for atomic segment-max --------
__device__ __forceinline__ unsigned f2ord(float f) {
  unsigned u = __float_as_uint(f);
  return (u & 0x80000000u) ? ~u : (u | 0x80000000u);
}
__device__ __forceinline__ float ord2f(unsigned u) {
  u = (u & 0x80000000u) ? (u & 0x7FFFFFFFu) : ~u;
  return __uint_as_float(u);
}

// ---------------------------------------------------------------------------
// Per-node: attention scalars a_src[i]=h[i]·aw_s, a_dst[i]=h[i]·aw_d,
// and init encoded running max to -inf.
// ---------------------------------------------------------------------------
__global__ void node_prep(const float* __restrict__ h,
                          const float* __restrict__ aw_s,
                          const float* __restrict__ aw_d,
                          float* __restrict__ asrc, float* __restrict__ adst,
                          unsigned* __restrict__ m_u, int Nn)
{
  const int i = blockIdx.x * blockDim.x + threadIdx.x;
  if (i >= Nn) return;
  const float4* hr = (const float4*)(h + (size_t)i * DOUT);
  const float4* ps = (const float4*)aw_s;
  const float4* pd = (const float4*)aw_d;
  float s = 0.f, d = 0.f;
#pragma unroll
  for (int q = 0; q < DOUT / 4; ++q) {
    float4 hv = hr[q], sv = ps[q], dv = pd[q];
    s += hv.x * sv.x + hv.y * sv.y + hv.z * sv.z + hv.w * sv.w;
    d += hv.x * dv.x + hv.y * dv.y + hv.z * dv.z + hv.w * dv.w;
  }
  asrc[i] = s;
  adst[i] = d;
  m_u[i] = 0x007FFFFFu;  // f2ord(-inf)
}

// ---------------------------------------------------------------------------
// Edge pass A: logit = leaky_relu(asrc[src]+adst[dst]); store; segment-max.
// Edges e>=E are the appended self-loops (src=dst=e-E).
// ---------------------------------------------------------------------------
__global__ void edge_logits_max(const int* __restrict__ src,
                                const int* __restrict__ dst,
                                int E, int Et,
                                const float* __restrict__ asrc,
                                const float* __restrict__ adst,
                                float* __restrict__ logit,
                                unsigned* __restrict__ m_u)
{
  const int e = blockIdx.x * blockDim.x + threadIdx.x;
  if (e >= Et) return;
  const int s = (e < E) ? src[e] : (e - E);
  const int d = (e < E) ? dst[e] : (e - E);
  const float t = asrc[s] + adst[d];
  const float lg = t > 0.f ? t : t * NEG_SLOPE;
  logit[e] = lg;
  atomicMax(&m_u[d], f2ord(lg));
}

// ---------------------------------------------------------------------------
// Edge pass B: p = exp(logit - m[dst]); ssum[dst] += p;
// acc[dst,:] += p * h[src,:]. One wave32 per edge, 2 features per lane.
// (Normalization by (s+eps) is deferred to the epilogue: mathematically
// identical to per-edge alpha and saves a full third edge sweep.)
// ---------------------------------------------------------------------------
__global__ __launch_bounds__(256) void edge_scatter(
    const int* __restrict__ src, const int* __restrict__ dst,
    int E, int Et,
    const float* __restrict__ logit, const unsigned* __restrict__ m_u,
    float* __restrict__ ssum,
    const float* __restrict__ h, float* __restrict__ acc)
{
  const int e    = (int)((blockIdx.x * (size_t)blockDim.x + threadIdx.x) >> 5);
  const int lane = threadIdx.x & 31;
  if (e >= Et) return;
  const int s = (e < E) ? src[e] : (e - E);
  const int d = (e < E) ? dst[e] : (e - E);
  const float p = __expf(logit[e] - ord2f(m_u[d]));
  if (lane == 0) atomicAdd(&ssum[d], p);
  const float2 hv = ((const float2*)(h + (size_t)s * DOUT))[lane];
  float* ao = acc + (size_t)d * DOUT + 2 * lane;
  atomicAdd(ao,     p * hv.x);
  atomicAdd(ao + 1, p * hv.y);
}

// ---------------------------------------------------------------------------
// Per-node epilogue: v = relu(acc/(s+eps) + bias); write back in-place;
// accumulate per-feature sum / sumsq for BatchNorm (block LDS + atomics).
// blockDim = (64, ROWS)
// ---------------------------------------------------------------------------
#define FIN_ROWS 8
__global__ __launch_bounds__(64 * FIN_ROWS) void finalize_stats(
    float* __restrict__ acc, const float* __restrict__ ssum,
    const float* __restrict__ bias, int Nn,
    float* __restrict__ colsum, float* __restrict__ colsumsq)
{
  __shared__ float cs[DOUT];
  __shared__ float cq[DOUT];
  const int j = threadIdx.x;                          // feature
  const int i = blockIdx.x * FIN_ROWS + threadIdx.y;  // node
  if (threadIdx.y == 0) { cs[j] = 0.f; cq[j] = 0.f; }
  __syncthreads();
  if (i < Nn) {
    float v = acc[(size_t)i * DOUT + j] / (ssum[i] + SM_EPS) + bias[j];
    v = v > 0.f ? v : 0.f;
    acc[(size_t)i * DOUT + j] = v;
    atomicAdd(&cs[j], v);
    atomicAdd(&cq[j], v * v);
  }
  __syncthreads();
  if (threadIdx.y == 0) {
    atomicAdd(&colsum[j], cs[j]);
    atomicAdd(&colsumsq[j], cq[j]);
  }
}

// ---------------------------------------------------------------------------
// BN stats (biased, training mode): scale = g*rsqrt(var+eps), shift = bb-mean*scale
// ---------------------------------------------------------------------------
__global__ void bn_stats(const float* __restrict__ colsum,
                         const float* __restrict__ colsumsq,
                         const float* __restrict__ g, const float* __restrict__ bb,
                         float* __restrict__ scale, float* __restrict__ shift, int Nn)
{
  const int j = threadIdx.x;
  const float inv = 1.0f / (float)Nn;
  const float mean = colsum[j] * inv;
  const float var = colsumsq[j] * inv - mean * mean;
  const float sc = g[j] * rsqrtf(var + BN_EPS);
  scale[j] = sc;
  shift[j] = bb[j] - mean * sc;
}

__global__ void bn_apply(const float* __restrict__ v,
                         const float* __restrict__ scale,
                         const float* __restrict__ shift,
                         float* __restrict__ y, size_t total)
{
  const size_t idx = blockIdx.x * (size_t)blockDim.x + threadIdx.x;
  if (idx >= total) return;
  const int j = (int)(idx & (DOUT - 1));
  y[idx] = v[idx] * scale[j] + shift[j];
}

// ---------------------------------------------------------------------------
// Host side
// ---------------------------------------------------------------------------
struct LayerBufs {
  float *h, *acc, *asrc, *adst, *ssum, *logit, *colsum, *colsumsq, *scale, *shift;
  unsigned* m_u;
};

static void run_layer(const float* X, int K,
                      const float* W, const float* aws, const float* awd,
                      const float* bias, const float* g, const float* bb,
                      const int* src, const int* dst, int E, int Et, int N,
                      float* out, const LayerBufs& B, hipStream_t stream)
{
  const size_t N64 = (size_t)N * DOUT;
  hipMemsetAsync(B.acc, 0, N64 * sizeof(float), stream);
  hipMemsetAsync(B.ssum, 0, (size_t)N * sizeof(float), stream);
  hipMemsetAsync(B.colsum, 0, 2 * DOUT * sizeof(float), stream);  // colsum+colsumsq contiguous

  if (K == 10) {
    gemm_wmma_f32<10><<<(N + 15) / 16, 128, 0, stream>>>(X, W, B.h, N);
  } else {
    gemm_wmma_f32<64><<<(N + 15) / 16, 128, 0, stream>>>(X, W, B.h, N);
  }
  node_prep<<<(N + 255) / 256, 256, 0, stream>>>(B.h, aws, awd, B.asrc, B.adst, B.m_u, N);
  edge_logits_max<<<(Et + 255) / 256, 256, 0, stream>>>(src, dst, E, Et, B.asrc, B.adst, B.logit, B.m_u);
  {
    size_t threads = (size_t)Et * 32;
    edge_scatter<<<(unsigned)((threads + 255) / 256), 256, 0, stream>>>(
        src, dst, E, Et, B.logit, B.m_u, B.ssum, B.h, B.acc);
  }
  finalize_stats<<<(N + FIN_ROWS - 1) / FIN_ROWS, dim3(DOUT, FIN_ROWS), 0, stream>>>(
      B.acc, B.ssum, bias, N, B.colsum, B.colsumsq);
  bn_stats<<<1, DOUT, 0, stream>>>(B.colsum, B.colsumsq, g, bb, B.scale, B.shift, N);
  bn_apply<<<(unsigned)((N64 + 255) / 256), 256, 0, stream>>>(B.acc, B.scale, B.shift, out, N64);
}

extern "C" void kernel_launch(void* const* d_in, const int* in_sizes, int n_in,
                              void* d_out, int out_size, void* d_ws, size_t ws_size,
                              hipStream_t stream)
{
  const float* x  = (const float*)d_in[0];
  const int* ei   = (const int*)d_in[1];
  // d_in[2] = edge_attr: accepted but unused (GATConv edge_dim=None)
  const float* W1  = (const float*)d_in[3];
  const float* as1 = (const float*)d_in[4];
  const float* ad1 = (const float*)d_in[5];
  const float* b1  = (const float*)d_in[6];
  const float* g1  = (const float*)d_in[7];
  const float* bb1 = (const float*)d_in[8];
  const float* W2  = (const float*)d_in[9];
  const float* as2 = (const float*)d_in[10];
  const float* ad2 = (const float*)d_in[11];
  const float* b2  = (const float*)d_in[12];
  const float* g2  = (const float*)d_in[13];
  const float* bb2 = (const float*)d_in[14];

  const int IN_DIM = 10;
  const int N  = in_sizes[0] / IN_DIM;
  const int E  = in_sizes[1] / 2;
  const int Et = E + N;  // with self-loops
  const int* src = ei;
  const int* dst = ei + E;

  const size_t N64 = (size_t)N * DOUT;
  float* ws = (float*)d_ws;
  LayerBufs B;
  B.h        = ws;                 ws += N64;
  B.acc      = ws;                 ws += N64;
  float* x2  = ws;                 ws += N64;   // layer-1 output / layer-2 input
  B.asrc     = ws;                 ws += N;
  B.adst     = ws;                 ws += N;
  B.m_u      = (unsigned*)ws;      ws += N;
  B.ssum     = ws;                 ws += N;
  B.logit    = ws;                 ws += Et;
  B.colsum   = ws;                 ws += DOUT;
  B.colsumsq = ws;                 ws += DOUT;
  B.scale    = ws;                 ws += DOUT;
  B.shift    = ws;                 ws += DOUT;

  // Layer 1: K = IN_DIM (10), Layer 2: K = 64
  run_layer(x,  IN_DIM, W1, as1, ad1, b1, g1, bb1, src, dst, E, Et, N, x2,            B, stream);
  run_layer(x2, DOUT,   W2, as2, ad2, b2, g2, bb2, src, dst, E, Et, N, (float*)d_out, B, stream);
}